// ROIPoolling_53575422050529
// MI455X (gfx1250) — compile-verified
//
#include <hip/hip_runtime.h>
#include <hip/hip_bf16.h>
#include <math.h>

// ---------------------------------------------------------------------------
// ROI max-pool over a 3-level feature pyramid (MI455X / gfx1250, wave32).
//
// Memory-bound gather-reduce (1 v_max per float). CDNA5 strategy:
//  * one block (8 wave32s) per output ROI slot, one wave per 32 channels
//  * the 2D ROI window is streamed into LDS by the Tensor Data Mover:
//    one tensor_load_to_lds per multi-row chunk (TDM walks the strided
//    rows itself), double-buffered per wave, tracked with TENSORcnt
//  * window == tile (always in-bounds), so consumption is a maskless
//    linear sweep of LDS via ds_load_b128; wave32 __shfl_xor max reduction
//  * global_prefetch_b8 pulls the next channel's window toward L2
// ---------------------------------------------------------------------------

#define IMG_W  1024.0f
#define IMG_H  800.0f
#define WAVES  8
#define NROI   512
#define NCH    256
#define BUFELT 512            // floats per TDM chunk buffer (2 KB)

typedef unsigned int v4u __attribute__((ext_vector_type(4)));
typedef int          v8i __attribute__((ext_vector_type(8)));
typedef int          v4i __attribute__((ext_vector_type(4)));
typedef float        v4f __attribute__((ext_vector_type(4)));

// ---------------- kernel 1: level assignment, window coords, stable perm ---

__global__ __launch_bounds__(NROI) void roi_meta_kernel(
    const float* __restrict__ rois, int* __restrict__ meta,
    float* __restrict__ idx_out, int B, int N)
{
    int b = blockIdx.x;
    int i = threadIdx.x;
    __shared__ int lvl[NROI];

    const float* r = rois + ((size_t)b * N + i) * 4;
    float rx1 = r[0], ry1 = r[1], rx2 = r[2], ry2 = r[3];
    float area = (rx2 - rx1) * (ry2 - ry1);
    int l = (area <= 2100.0f) ? 0 : ((area < 10575.0f) ? 1 : 2);
    lvl[i] = l;
    __syncthreads();

    // stable rank of roi i within (level band, original order)
    int rank = 0;
    for (int j = 0; j < N; ++j) {
        int lj = lvl[j];
        rank += (int)((lj < l) | ((lj == l) & (j < i)));
    }

    int w = (l == 0) ? 256 : ((l == 1) ? 128 : 64);
    int h = (l == 0) ? 200 : ((l == 1) ? 100 : 50);

    int x1 = (int)fmaxf(floorf(rx1 * (1.0f / IMG_W) * (float)w), 0.0f);
    int y1 = (int)fmaxf(floorf(ry1 * (1.0f / IMG_H) * (float)h), 0.0f);
    int x2 = (int)fmaxf(ceilf (rx2 * (1.0f / IMG_W) * (float)w), 0.0f);
    int y2 = (int)fmaxf(ceilf (ry2 * (1.0f / IMG_H) * (float)h), 0.0f);
    if (x1 == 0 && x2 == 0) x2 = 1;
    if (y1 == 0 && y2 == 0) y2 = 1;
    if (x1 >= w) x1 = w - 1;
    if (y1 >= h) y1 = h - 1;

    int slot = b * N + rank;
    int* m = meta + (size_t)slot * 6;
    m[0] = l; m[1] = x1; m[2] = x2; m[3] = y1; m[4] = y2; m[5] = i;
    // reference's (off-by-one) flattened index: perm + b*(N-1)
    idx_out[slot] = (float)(i + b * (N - 1));
}

// ---------------- TDM chunk issue ------------------------------------------

__device__ __forceinline__ unsigned rfl(unsigned x)
{
    return (unsigned)__builtin_amdgcn_readfirstlane((int)x);
}

// Issue one TDM 2D-tile load: rows x tw floats, row stride w, into LDS at
// lds_off. All descriptor fields are wave-uniform -> forced to SGPRs.
__device__ __forceinline__ void tdm_issue(const float* __restrict__ gsrc,
                                          unsigned lds_off,
                                          int rows, int tw, int w)
{
    // close the WAR race: all prior ds_loads from this buffer must be done
    // before the TDM may overwrite it
    asm volatile("s_wait_dscnt 0" ::: "memory");

    unsigned long long ga = (unsigned long long)(size_t)gsrc;

    v4u g0;
    g0.x = rfl(1u);                                   // count=1 (valid), user
    g0.y = rfl(lds_off);                              // lds_addr (bytes)
    g0.z = rfl((unsigned)ga);                         // global_addr[31:0]
    g0.w = rfl(((unsigned)(ga >> 32) & 0x1FFFFFFu)    // global_addr[56:32]
               | (2u << 30));                         // type = 2 ("image")

    unsigned utw = (unsigned)tw, urows = (unsigned)rows, uw = (unsigned)w;
    v8i g1;
    g1[0] = (int)rfl(2u << 16);                       // wg_mask=0, data_size=4B
    g1[1] = (int)rfl((utw & 0xFFFFu) << 16);          // tensor_dim0 lo16
    g1[2] = (int)rfl(((utw >> 16) & 0xFFFFu)          // tensor_dim0 hi16
               | ((urows & 0xFFFFu) << 16));          // tensor_dim1 lo16
    g1[3] = (int)rfl(((urows >> 16) & 0xFFFFu)        // tensor_dim1 hi16
               | ((utw & 0xFFFFu) << 16));            // tile_dim0 = tw
    g1[4] = (int)rfl(urows & 0xFFFFu);                // tile_dim1 = rows, tile_dim2=0
    g1[5] = (int)rfl(uw);                             // tensor_dim0_stride lo32 = w
    g1[6] = (int)rfl(0u);                             // stride hi16 | dim1_stride lo16
    g1[7] = (int)rfl(0u);                             // dim1_stride hi32 (2D: unused)

    v4i z4 = {0, 0, 0, 0};                            // groups 2/3 unused (2D tile)
    v8i z8 = {0, 0, 0, 0, 0, 0, 0, 0};                // trailing group (clang-23 form)
    __builtin_amdgcn_tensor_load_to_lds(g0, g1, z4, z4, z8, 0);
}

// ---------------- kernel 2: TDM-pipelined max pool -------------------------

__global__ __launch_bounds__(WAVES * 32) void roi_pool_kernel(
    const float* __restrict__ f0, const float* __restrict__ f1,
    const float* __restrict__ f2, const int* __restrict__ meta,
    float* __restrict__ out, int B, int N, int C)
{
    __shared__ float stage[WAVES][2][BUFELT];   // per-wave double buffer (32 KB)

    int slot = blockIdx.x;
    int b    = slot / N;
    const int* m = meta + (size_t)slot * 6;
    int l = m[0], x1 = m[1], x2 = m[2], y1 = m[3], y2 = m[4];

    const float* feat = (l == 0) ? f0 : ((l == 1) ? f1 : f2);
    int w = (l == 0) ? 256 : ((l == 1) ? 128 : 64);
    int h = (l == 0) ? 200 : ((l == 1) ? 100 : 50);

    int wave = (int)threadIdx.x >> 5;
    int lane = (int)threadIdx.x & 31;

    int tw = x2 - x1;                 // window width  (1..64), fully in-bounds
    int th = y2 - y1;                 // window height (1..50), fully in-bounds
    int rpc = BUFELT / tw;            // rows per chunk
    if (rpc > th) rpc = th;
    int nChunks = (th + rpc - 1) / rpc;

    unsigned lds0 = (unsigned)(size_t)&stage[wave][0][0];
    unsigned lds1 = (unsigned)(size_t)&stage[wave][1][0];

    const float* bbase = feat + (size_t)b * C * h * w;

    for (int c = wave; c < C; c += WAVES) {
        const float* base = bbase + (size_t)c * h * w + (size_t)y1 * w + x1;
        // pull the next channel's window toward L2 while we stream this one
        if (c + WAVES < C)
            __builtin_prefetch(bbase + (size_t)(c + WAVES) * h * w
                               + (size_t)y1 * w + x1 + lane, 0, 0);

        float v = -INFINITY;

        // prologue: stage chunk 0
        tdm_issue(base, lds0, (nChunks == 1) ? th : rpc, tw, w);

        int kb = 0;
        for (int k = 0; k < nChunks; ++k) {             // wave-uniform loop
            if (k + 1 < nChunks) {
                int r1 = th - (k + 1) * rpc;
                if (r1 > rpc) r1 = rpc;
                tdm_issue(base + (size_t)(k + 1) * rpc * w,
                          (kb ^ 1) ? lds1 : lds0, r1, tw, w);
                asm volatile("s_wait_tensorcnt 0x1" ::: "memory");
            } else {
                asm volatile("s_wait_tensorcnt 0x0" ::: "memory");
            }
            int rows = th - k * rpc;
            if (rows > rpc) rows = rpc;
            int n = rows * tw;                          // tile == window: no mask
            const float* cur = &stage[wave][kb][0];

            // bulk sweep: ds_load_b128 (buffer is 16B-aligned)
            int n4 = n >> 2;
            const v4f* cur4 = (const v4f*)cur;
            for (int i = lane; i < n4; i += 32) {
                v4f q = cur4[i];
                v = fmaxf(v, fmaxf(fmaxf(q.x, q.y), fmaxf(q.z, q.w)));
            }
            // scalar tail (n % 4 elements)
            for (int i = (n4 << 2) + lane; i < n; i += 32)
                v = fmaxf(v, cur[i]);

            kb ^= 1;
        }

        // wave32 max reduction
#pragma unroll
        for (int off = 16; off > 0; off >>= 1)
            v = fmaxf(v, __shfl_xor(v, off, 32));

        if (lane == 0) out[(size_t)slot * C + c] = v;
    }
}

// ---------------- launcher -------------------------------------------------

extern "C" void kernel_launch(void* const* d_in, const int* in_sizes, int n_in,
                              void* d_out, int out_size, void* d_ws, size_t ws_size,
                              hipStream_t stream)
{
    const float* f0   = (const float*)d_in[0];   // [2,256,200,256]
    const float* f1   = (const float*)d_in[1];   // [2,256,100,128]
    const float* f2   = (const float*)d_in[2];   // [2,256, 50, 64]
    const float* rois = (const float*)d_in[3];   // [2,512,4]

    const int B = 2, N = NROI, C = NCH;

    float* out     = (float*)d_out;                       // [B,N,C]
    float* idx_out = out + (size_t)B * N * C;             // [B*N]
    int*   meta    = (int*)d_ws;                          // [B*N][6]

    roi_meta_kernel<<<B, N, 0, stream>>>(rois, meta, idx_out, B, N);
    roi_pool_kernel<<<B * N, WAVES * 32, 0, stream>>>(f0, f1, f2, meta, out, B, N, C);
}